// GraphEndNetwork_18322330485508
// MI455X (gfx1250) — compile-verified
//
#include <hip/hip_runtime.h>
#include <math.h>

typedef __attribute__((ext_vector_type(4))) float v4f;
typedef __attribute__((ext_vector_type(2))) float v2f;
typedef __attribute__((ext_vector_type(8))) float v8f;

#define N_NODES 8192
#define EMBED   64
#define H0      256
#define H1      256
#define H2      128
#define ACTION  8192
#define IN_SIZE (N_NODES + EMBED)   // 8256

// ---------------------------------------------------------------------------
// 1) diag[j] = state[j,j]
__global__ void k_diag(const float* __restrict__ state, float* __restrict__ diag) {
    int j = blockIdx.x * blockDim.x + threadIdx.x;
    if (j < N_NODES) diag[j] = state[(size_t)j * (N_NODES + 1)];
}

// ---------------------------------------------------------------------------
// 2) a[i] = sum_j state[i,j]*diag[j]  - diag[i]^2   (adj @ x with zeroed diag)
//    One block per row, fully coalesced float4 NT loads (state >> L2).
__global__ void k_rowdot(const float* __restrict__ state,
                         const float* __restrict__ diag,
                         float* __restrict__ a) {
    __shared__ float red[256];
    const int i   = blockIdx.x;
    const int tid = threadIdx.x;
    const v4f* row = (const v4f*)(state + (size_t)i * N_NODES);
    const v4f* dg  = (const v4f*)diag;
    float acc = 0.f;
#pragma unroll
    for (int it = 0; it < 8; ++it) {               // 8 * 256 * 4 = 8192 elems
        int idx = it * 256 + tid;
        v4f s = __builtin_nontemporal_load(row + idx);
        v4f d = dg[idx];
        acc += s.x * d.x + s.y * d.y + s.z * d.z + s.w * d.w;
    }
    red[tid] = acc;
    __syncthreads();
    for (int off = 128; off > 0; off >>= 1) {
        if (tid < off) red[tid] += red[tid + off];
        __syncthreads();
    }
    if (tid == 0) {
        float di = diag[i];
        a[i] = red[0] - di * di;                   // exclude diagonal term
    }
}

// ---------------------------------------------------------------------------
// 3) partial t: t[k] = sum_j w_j * relu(a_j*gc1_w[k] + gc1_b[k]),
//    w_j = state[start, j] (0 at j == start). 32 blocks, each owns 256 j's,
//    writes its own partial vector (deterministic, no float atomics).
__global__ void k_partial_t(const float* __restrict__ state,
                            const float* __restrict__ a,
                            const int*   __restrict__ start_p,
                            const float* __restrict__ gc1_w,
                            const float* __restrict__ gc1_b,
                            float* __restrict__ part_t) {
    __shared__ float aL[256];
    __shared__ float wL[256];
    const int tid = threadIdx.x;
    const int b   = blockIdx.x;
    const int j   = b * 256 + tid;
    const int start = start_p[0];
    aL[tid] = a[j];
    float w = state[(size_t)start * N_NODES + j];
    wL[tid] = (j == start) ? 0.f : w;
    __syncthreads();
    const float g1w = gc1_w[tid];
    const float g1b = gc1_b[tid];
    float acc = 0.f;
#pragma unroll 8
    for (int jj = 0; jj < 256; ++jj) {
        float h = fmaf(aL[jj], g1w, g1b);
        h = h > 0.f ? h : 0.f;
        acc = fmaf(wL[jj], h, acc);
    }
    part_t[b * 256 + tid] = acc;
}

// ---------------------------------------------------------------------------
// 4) reduce partials -> t[256]; s = t@gc2_w + gc2_b; relu; log_softmax -> emb[64]
__global__ void k_embed(const float* __restrict__ part_t,
                        const float* __restrict__ gc2_w,
                        const float* __restrict__ gc2_b,
                        float* __restrict__ emb) {
    __shared__ float tS[256];
    __shared__ float hS[64];
    __shared__ float lseS;
    const int tid = threadIdx.x;
    float acc = 0.f;
#pragma unroll
    for (int b = 0; b < 32; ++b) acc += part_t[b * 256 + tid];
    tS[tid] = acc;
    __syncthreads();
    if (tid < EMBED) {
        float s = gc2_b[tid];
        for (int k = 0; k < H0; ++k) s = fmaf(tS[k], gc2_w[k * EMBED + tid], s);
        hS[tid] = s > 0.f ? s : 0.f;               // relu BEFORE log_softmax
    }
    __syncthreads();
    if (tid == 0) {
        float mx = hS[0];
        for (int c = 1; c < EMBED; ++c) mx = fmaxf(mx, hS[c]);
        float se = 0.f;
        for (int c = 0; c < EMBED; ++c) se += expf(hS[c] - mx);
        lseS = mx + logf(se);
    }
    __syncthreads();
    if (tid < EMBED) emb[tid] = hS[tid] - lseS;
}

// ---------------------------------------------------------------------------
// 5) fc1: y1[r] = relu( fc1_w[r,:] . concat(state[start,:], emb) + fc1_b[r] )
//    One block per output row r (256 blocks).
__global__ void k_fc1(const float* __restrict__ state,
                      const float* __restrict__ emb,
                      const int*   __restrict__ start_p,
                      const float* __restrict__ fc1_w,
                      const float* __restrict__ fc1_b,
                      float* __restrict__ y1) {
    __shared__ float red[256];
    const int r   = blockIdx.x;
    const int tid = threadIdx.x;
    const int start = start_p[0];
    const float* wrow = fc1_w + (size_t)r * IN_SIZE;
    const float* srow = state + (size_t)start * N_NODES;
    float acc = 0.f;
    for (int idx = tid; idx < IN_SIZE; idx += 256) {
        float e = (idx < N_NODES) ? srow[idx] : emb[idx - N_NODES];
        acc = fmaf(wrow[idx], e, acc);
    }
    red[tid] = acc;
    __syncthreads();
    for (int off = 128; off > 0; off >>= 1) {
        if (tid < off) red[tid] += red[tid + off];
        __syncthreads();
    }
    if (tid == 0) {
        float v = red[0] + fc1_b[r];
        y1[r] = v > 0.f ? v : 0.f;
    }
}

// ---------------------------------------------------------------------------
// 6) fc2 + fc3 fused in a single block (tiny matvecs)
__global__ void k_fc23(const float* __restrict__ y1,
                       const float* __restrict__ fc2_w, const float* __restrict__ fc2_b,
                       const float* __restrict__ fc3_w, const float* __restrict__ fc3_b,
                       float* __restrict__ y3) {
    __shared__ float y1S[256];
    __shared__ float y2S[256];
    const int tid = threadIdx.x;
    y1S[tid] = y1[tid];
    __syncthreads();
    {
        const float* wr = fc2_w + (size_t)tid * H0;
        float acc = fc2_b[tid];
        for (int k = 0; k < H0; ++k) acc = fmaf(wr[k], y1S[k], acc);
        y2S[tid] = acc > 0.f ? acc : 0.f;
    }
    __syncthreads();
    if (tid < H2) {
        const float* wr = fc3_w + (size_t)tid * H1;
        float acc = fc3_b[tid];
        for (int k = 0; k < H1; ++k) acc = fmaf(wr[k], y2S[k], acc);
        y3[tid] = acc > 0.f ? acc : 0.f;
    }
}

// ---------------------------------------------------------------------------
// 7) fc4 via V_WMMA_F32_16X16X4_F32: each wave computes a 16-row tile of
//    out = relu(fc4_w @ y3 + fc4_b). y3 occupies column N=0 of B; we loop
//    K = 128 in steps of 4 (32 WMMAs/wave) and read column 0 of D.
//    A layout (ISA): lanes 0-15 hold {K0,K1} of row M=lane; lanes 16-31 hold
//    {K2,K3} of row M=lane-16. D column 0 lives in lane 0 (M=0..7) and
//    lane 16 (M=8..15), VGPRs 0..7.
__global__ void k_fc4_wmma(const float* __restrict__ y3,
                           const float* __restrict__ fc4_w,
                           const float* __restrict__ fc4_b,
                           float* __restrict__ out) {
    const int gtid = blockIdx.x * blockDim.x + threadIdx.x;
    const int wave = gtid >> 5;          // 512 waves -> 512 tiles of 16 rows
    const int lane = gtid & 31;
    const int lrow = lane & 15;
    const int hi   = (lane >= 16) ? 1 : 0;
    const int rowbase = wave * 16;
    const float* wrow = fc4_w + (size_t)(rowbase + lrow) * H2 + (hi ? 2 : 0);

    v8f c = {};
#pragma unroll
    for (int kk = 0; kk < 32; ++kk) {
        const int k0 = kk * 4;
        v2f av;
        av.x = wrow[k0];
        av.y = wrow[k0 + 1];
        // B: column 0 = y3 slice, all other columns zero (lanes with lrow!=0)
        float b0 = y3[k0 + (hi ? 2 : 0)];
        float b1 = y3[k0 + (hi ? 3 : 1)];
        v2f bv;
        bv.x = (lrow == 0) ? b0 : 0.f;
        bv.y = (lrow == 0) ? b1 : 0.f;
        c = __builtin_amdgcn_wmma_f32_16x16x4_f32(
                /*neg_a=*/false, av, /*neg_b=*/false, bv,
                /*c_mod=*/(short)0, c, /*reuse_a=*/false, /*reuse_b=*/false);
    }
    if (lrow == 0) {
        const int mbase = rowbase + (hi ? 8 : 0);
#pragma unroll
        for (int p = 0; p < 8; ++p) {
            float v = c[p] + fc4_b[mbase + p];
            out[mbase + p] = v > 0.f ? v : 0.f;
        }
    }
}

// ---------------------------------------------------------------------------
extern "C" void kernel_launch(void* const* d_in, const int* in_sizes, int n_in,
                              void* d_out, int out_size, void* d_ws, size_t ws_size,
                              hipStream_t stream) {
    const float* state  = (const float*)d_in[0];
    const int*   startp = (const int*)  d_in[1];
    const float* gc1_w  = (const float*)d_in[2];
    const float* gc1_b  = (const float*)d_in[3];
    const float* gc2_w  = (const float*)d_in[4];
    const float* gc2_b  = (const float*)d_in[5];
    const float* fc1_w  = (const float*)d_in[6];
    const float* fc1_b  = (const float*)d_in[7];
    const float* fc2_w  = (const float*)d_in[8];
    const float* fc2_b  = (const float*)d_in[9];
    const float* fc3_w  = (const float*)d_in[10];
    const float* fc3_b  = (const float*)d_in[11];
    const float* fc4_w  = (const float*)d_in[12];
    const float* fc4_b  = (const float*)d_in[13];
    float* out = (float*)d_out;

    float* ws     = (float*)d_ws;
    float* diag   = ws;                      // 8192
    float* a      = ws + 8192;               // 8192
    float* part_t = ws + 16384;              // 32*256 = 8192
    float* emb    = ws + 24576;              // 64
    float* y1     = emb + 64;                // 256
    float* y3     = y1 + 256;                // 128

    k_diag     <<<N_NODES / 256, 256, 0, stream>>>(state, diag);
    k_rowdot   <<<N_NODES,       256, 0, stream>>>(state, diag, a);
    k_partial_t<<<32,            256, 0, stream>>>(state, a, startp, gc1_w, gc1_b, part_t);
    k_embed    <<<1,             256, 0, stream>>>(part_t, gc2_w, gc2_b, emb);
    k_fc1      <<<H0,            256, 0, stream>>>(state, emb, startp, fc1_w, fc1_b, y1);
    k_fc23     <<<1,             256, 0, stream>>>(y1, fc2_w, fc2_b, fc3_w, fc3_b, y3);
    k_fc4_wmma <<<ACTION / (8 * 16), 256, 0, stream>>>(y3, fc4_w, fc4_b, out);
}